// TopicAttentiveEmbedding_44590350467081
// MI455X (gfx1250) — compile-verified
//
#include <hip/hip_runtime.h>

typedef __attribute__((ext_vector_type(2))) float v2f;
typedef __attribute__((ext_vector_type(4))) float v4f;
typedef __attribute__((ext_vector_type(8))) float v8f;

#define BB 64
#define SS 512
#define DD 768
#define CHUNKS 8                 // s-chunks per batch for the scores kernel
#define ROWS_PER_CHUNK (SS / CHUNKS)   // 64

// ---------------------------------------------------------------------------
// Kernel 1: tW[64,768] = topic[64,768] @ W[768,768] via V_WMMA_F32_16X16X4_F32
// One wave32 per 16x16 output tile. Grid = (48 N-tiles, 4 M-tiles).
// ---------------------------------------------------------------------------
__global__ __launch_bounds__(32) void tw_gemm_wmma_f32(
    const float* __restrict__ topic,
    const float* __restrict__ W,
    float* __restrict__ tW)
{
  const int tileN = blockIdx.x;        // 0..47
  const int tileM = blockIdx.y;        // 0..3
  const int lane  = threadIdx.x;       // 0..31
  const int half  = lane >> 4;         // 0 | 1
  const int l16   = lane & 15;

  const int m = tileM * 16 + l16;      // A row held by this lane
  const int n = tileN * 16 + l16;      // B column held by this lane
  const float* __restrict__ arow = topic + m * DD;

  v8f c = {};
  for (int k = 0; k < DD; k += 4) {
    const int kk = k + 2 * half;       // lanes 0-15: K={k,k+1}; 16-31: K={k+2,k+3}
    v2f a;
    a.x = arow[kk];
    a.y = arow[kk + 1];
    v2f b;
    b.x = W[(size_t)kk * DD + n];
    b.y = W[(size_t)(kk + 1) * DD + n];
    c = __builtin_amdgcn_wmma_f32_16x16x4_f32(
        /*neg_a=*/false, a, /*neg_b=*/false, b,
        /*c_mod=*/(short)0, c, /*reuse_a=*/false, /*reuse_b=*/false);
  }

  // 16x16 f32 C/D layout: VGPR r holds M = r + 8*half, N = l16
#pragma unroll
  for (int r = 0; r < 8; ++r) {
    const int mm = tileM * 16 + r + 8 * half;
    tW[(size_t)mm * DD + tileN * 16 + l16] = c[r];
  }
}

// ---------------------------------------------------------------------------
// Kernel 2: scores[b,s] = tW[b].seq[b,s] + bias. HBM-bound streaming stage.
// Grid = (CHUNKS, BB) = 512 blocks x 256 threads (8 wave32) for occupancy.
// Each wave processes 2 rows at a time (dual accumulators -> 12 b128 loads
// in flight per wave per step), fully coalesced 512B/step per wave.
// ---------------------------------------------------------------------------
__global__ __launch_bounds__(256) void scores_kernel(
    const float* __restrict__ tW,
    const float* __restrict__ seq,
    const float* __restrict__ bias,
    float* __restrict__ scores)
{
  __shared__ float s_tw[DD];

  const int b     = blockIdx.y;
  const int chunk = blockIdx.x;        // 0..CHUNKS-1
  const int tid   = threadIdx.x;
  const int lane  = tid & 31;
  const int wave  = tid >> 5;          // 0..7

  for (int i = tid; i < DD; i += 256)
    s_tw[i] = tW[(size_t)b * DD + i];
  __syncthreads();

  const float* __restrict__ seqb =
      seq + ((size_t)b * SS + chunk * ROWS_PER_CHUNK) * DD;
  float* __restrict__ scb = scores + (size_t)b * SS + chunk * ROWS_PER_CHUNK;
  const float bi = bias[0];

  const int r0 = wave * (ROWS_PER_CHUNK / 8);      // 8 rows per wave
  const int r1 = r0 + (ROWS_PER_CHUNK / 8);

  for (int r = r0; r < r1; r += 2) {
    const float* __restrict__ rowA = seqb + (size_t)r * DD;
    const float* __restrict__ rowB = rowA + DD;

    // speculative prefetch two rows ahead (global_prefetch_b8)
    if (r + 3 < ROWS_PER_CHUNK) {
      __builtin_prefetch(rowA + 2 * DD + lane * 16, 0, 0);
      __builtin_prefetch(rowB + 2 * DD + lane * 16, 0, 0);
    }

    float a0 = 0.f, a1 = 0.f;
#pragma unroll
    for (int ch = 0; ch < DD / (32 * 4); ++ch) {
      const int idx = (ch * 32 + lane) * 4;
      const v4f x0 = *(const v4f*)(rowA + idx);
      const v4f x1 = *(const v4f*)(rowB + idx);
      const v4f w4 = *(const v4f*)(&s_tw[idx]);
      a0 += x0.x * w4.x + x0.y * w4.y + x0.z * w4.z + x0.w * w4.w;
      a1 += x1.x * w4.x + x1.y * w4.y + x1.z * w4.z + x1.w * w4.w;
    }
#pragma unroll
    for (int off = 16; off > 0; off >>= 1) {
      a0 += __shfl_xor(a0, off, 32);
      a1 += __shfl_xor(a1, off, 32);
    }
    if (lane == 0) {
      scb[r]     = a0 + bi;
      scb[r + 1] = a1 + bi;
    }
  }
}

// ---------------------------------------------------------------------------
// Kernel 3: softmax over S=512 per batch. 64 blocks x 512 threads; touches
// only 128 KB total -> negligible.
// ---------------------------------------------------------------------------
__global__ __launch_bounds__(512) void softmax_kernel(
    const float* __restrict__ scores,
    float* __restrict__ out)
{
  __shared__ float s_red[16];

  const int b    = blockIdx.x;
  const int tid  = threadIdx.x;
  const int lane = tid & 31;
  const int wave = tid >> 5;           // 0..15

  const float v = scores[(size_t)b * SS + tid];

  float mx = v;
#pragma unroll
  for (int off = 16; off > 0; off >>= 1)
    mx = fmaxf(mx, __shfl_xor(mx, off, 32));
  if (lane == 0) s_red[wave] = mx;
  __syncthreads();
  if (wave == 0) {
    float t = s_red[lane & 15];
#pragma unroll
    for (int off = 8; off > 0; off >>= 1)
      t = fmaxf(t, __shfl_xor(t, off, 32));
    if (lane == 0) s_red[0] = t;
  }
  __syncthreads();
  mx = s_red[0];

  const float e = __expf(v - mx);

  float sum = e;
#pragma unroll
  for (int off = 16; off > 0; off >>= 1)
    sum += __shfl_xor(sum, off, 32);
  __syncthreads();                      // s_red reuse hazard
  if (lane == 0) s_red[wave] = sum;
  __syncthreads();
  if (wave == 0) {
    float t = s_red[lane & 15];
#pragma unroll
    for (int off = 8; off > 0; off >>= 1)
      t += __shfl_xor(t, off, 32);
    if (lane == 0) s_red[0] = t;
  }
  __syncthreads();
  sum = s_red[0];

  out[(size_t)b * SS + tid] = e / sum;
}

// ---------------------------------------------------------------------------
extern "C" void kernel_launch(void* const* d_in, const int* in_sizes, int n_in,
                              void* d_out, int out_size, void* d_ws, size_t ws_size,
                              hipStream_t stream) {
  (void)in_sizes; (void)n_in; (void)out_size; (void)ws_size;

  const float* topic = (const float*)d_in[0];   // [64,768]
  const float* seq   = (const float*)d_in[1];   // [64,512,768]
  const float* W     = (const float*)d_in[2];   // [768,768]
  const float* bias  = (const float*)d_in[3];   // [1]
  float* out = (float*)d_out;                   // [64,512]

  float* tW     = (float*)d_ws;                 // 64*768 f32   (192 KiB)
  float* scores = tW + (size_t)BB * DD;         // 64*512 f32   (128 KiB)

  dim3 g1(DD / 16, BB / 16);                    // (48, 4) tiles
  tw_gemm_wmma_f32<<<g1, 32, 0, stream>>>(topic, W, tW);

  dim3 g2(CHUNKS, BB);                          // (8, 64) = 512 blocks
  scores_kernel<<<g2, 256, 0, stream>>>(tW, seq, bias, scores);

  softmax_kernel<<<BB, SS, 0, stream>>>(scores, out);
}